// TripletLoss_44006234915136
// MI455X (gfx1250) — compile-verified
//
#include <hip/hip_runtime.h>
#include <math.h>

typedef __attribute__((ext_vector_type(2))) float v2f;
typedef __attribute__((ext_vector_type(8))) float v8f;

#define NPTS 512
#define DIM  2048
#define MARGIN_F 0.5f

// ---------------------------------------------------------------------------
// Gram matrix G = x * x^T via V_WMMA_F32_16X16X4_F32 (fp32, full precision).
// One wave per 16x16 output tile; 32x32 = 1024 tiles.
// A-frag / B-frag per-lane layout (ISA 7.12.2, 32-bit A 16x4):
//   lane L: row = L&15, k-pair offset = 2*(L>>4)  -> contiguous float2 load.
// ---------------------------------------------------------------------------
__global__ void gram_wmma_kernel(const float* __restrict__ x,
                                 float* __restrict__ g) {
    const int lane = threadIdx.x & 31;
    const int wave = threadIdx.x >> 5;
    const int tile = blockIdx.x * (blockDim.x >> 5) + wave;   // 0..1023
    const int rbase = (tile >> 5) << 4;   // tile row * 16
    const int cbase = (tile & 31) << 4;   // tile col * 16
    const int m    = lane & 15;
    const int koff = (lane >> 4) << 1;    // 0 or 2

    const float* pa = x + (size_t)(rbase + m) * DIM + koff;
    const float* pb = x + (size_t)(cbase + m) * DIM + koff;

    v8f c0 = {};
    v8f c1 = {};
    for (int k = 0; k < DIM; k += 8) {
        v2f a0 = *(const v2f*)(pa + k);
        v2f b0 = *(const v2f*)(pb + k);
        v2f a1 = *(const v2f*)(pa + k + 4);
        v2f b1 = *(const v2f*)(pb + k + 4);
        c0 = __builtin_amdgcn_wmma_f32_16x16x4_f32(false, a0, false, b0,
                                                   (short)0, c0, false, false);
        c1 = __builtin_amdgcn_wmma_f32_16x16x4_f32(false, a1, false, b1,
                                                   (short)0, c1, false, false);
    }
    v8f c = c0 + c1;

    // C/D layout: VGPR r -> row (r + 8*(lane>>4)), col = lane&15
    const int col = cbase + m;
    const int row0 = rbase + ((lane >> 4) << 3);
#pragma unroll
    for (int r = 0; r < 8; ++r)
        g[(size_t)(row0 + r) * NPTS + col] = c[r];
}

// ---------------------------------------------------------------------------
// One block per anchor `a`. Builds dist row + same-label mask in LDS from the
// Gram row and diagonal, then accumulates violating triplet margins.
//   viol(p,n) = same[p] & p!=a & !same[n] & (d_ap - d_an + M > 0)
// ---------------------------------------------------------------------------
__global__ void triplet_kernel(const float* __restrict__ g,
                               const int* __restrict__ tgt,
                               float* __restrict__ acc_sum,
                               unsigned int* __restrict__ acc_cnt) {
    __shared__ float         s_dist[NPTS];
    __shared__ unsigned char s_same[NPTS];
    __shared__ float         s_rs[256];
    __shared__ unsigned int  s_rc[256];

    const int a   = blockIdx.x;
    const int tid = threadIdx.x;
    const int ta  = tgt[a];
    const float gaa = g[(size_t)a * NPTS + a];

    for (int j = tid; j < NPTS; j += blockDim.x) {
        float gjj = g[(size_t)j * NPTS + j];
        float gaj = g[(size_t)a * NPTS + j];
        float d2  = gaa + gjj - 2.0f * gaj;
        s_dist[j] = (d2 > 0.0f) ? sqrtf(d2) : 0.0f;
        s_same[j] = (tgt[j] == ta) ? 1 : 0;
    }
    __syncthreads();

    float        sum = 0.0f;
    unsigned int cnt = 0;
    for (int p = 0; p < NPTS; ++p) {
        if (p == a || !s_same[p]) continue;          // positives only (~8/anchor)
        const float dap = s_dist[p];
        for (int n = tid; n < NPTS; n += blockDim.x) {
            if (s_same[n]) continue;                 // negatives only
            float per = dap - s_dist[n] + MARGIN_F;
            if (per > 0.0f) { sum += per; cnt++; }   // strict: d_an - d_ap < M
        }
    }

    s_rs[tid] = sum;
    s_rc[tid] = cnt;
    __syncthreads();
    for (int s = blockDim.x >> 1; s > 0; s >>= 1) {
        if (tid < s) {
            s_rs[tid] += s_rs[tid + s];
            s_rc[tid] += s_rc[tid + s];
        }
        __syncthreads();
    }
    if (tid == 0) {
        atomicAdd(acc_sum, s_rs[0]);
        atomicAdd(acc_cnt, s_rc[0]);
    }
}

__global__ void init_acc_kernel(float* acc_sum, unsigned int* acc_cnt) {
    *acc_sum = 0.0f;
    *acc_cnt = 0u;
}

__global__ void finalize_kernel(const float* __restrict__ acc_sum,
                                const unsigned int* __restrict__ acc_cnt,
                                float* __restrict__ out) {
    unsigned int c = *acc_cnt;
    if (c < 1u) c = 1u;
    out[0] = (*acc_sum) / (float)c;
}

// ---------------------------------------------------------------------------
// Workspace layout: [0, 1MB) Gram matrix (512*512 f32), then sum + cnt.
// ---------------------------------------------------------------------------
extern "C" void kernel_launch(void* const* d_in, const int* in_sizes, int n_in,
                              void* d_out, int out_size, void* d_ws, size_t ws_size,
                              hipStream_t stream) {
    const float* x   = (const float*)d_in[0];     // (512, 2048) f32
    const int*   tgt = (const int*)d_in[1];       // (512,) labels

    float*        g       = (float*)d_ws;
    float*        acc_sum = g + (size_t)NPTS * NPTS;
    unsigned int* acc_cnt = (unsigned int*)(acc_sum + 1);

    init_acc_kernel<<<1, 1, 0, stream>>>(acc_sum, acc_cnt);
    // 1024 tiles / 8 waves per 256-thread block -> 128 blocks
    gram_wmma_kernel<<<128, 256, 0, stream>>>(x, g);
    triplet_kernel<<<NPTS, 256, 0, stream>>>(g, tgt, acc_sum, acc_cnt);
    finalize_kernel<<<1, 1, 0, stream>>>(acc_sum, acc_cnt, (float*)d_out);
}